// TopKRouter_10222022165062
// MI455X (gfx1250) — compile-verified
//
#include <hip/hip_runtime.h>
#include <math.h>

// Problem constants (match reference)
#define T_TOKENS 16384
#define DIM      2048
#define NE       64      // num experts
#define TOPK     8
// Tiling
#define KC       128     // K-chunk staged in LDS
#define BT       64      // tokens per block
#define PAD      4       // LDS row padding (floats) -> bank-conflict-free fragment reads
#define XROW     (KC + PAD)
#define SROW     (NE + PAD)

typedef float v2f __attribute__((ext_vector_type(2)));
typedef float v8f __attribute__((ext_vector_type(8)));

#ifndef __has_builtin
#define __has_builtin(x) 0
#endif

// CDNA5 async global->LDS copy path (ASYNCcnt); falls back to sync copies on
// the host pass / toolchains without the builtin.
#if __has_builtin(__builtin_amdgcn_global_load_async_to_lds_b128)
#define USE_ASYNC 1
typedef int v4i __attribute__((vector_size(16)));
typedef __attribute__((address_space(1))) v4i gv4i;   // global (AS1) int4
typedef __attribute__((address_space(3))) v4i lv4i;   // LDS (AS3) int4

__device__ __forceinline__ void async_copy_b128(const void* g, void* l) {
  // AS3 pointer = low 32 bits of the generic LDS address (ISA 10.2 aperture)
  __builtin_amdgcn_global_load_async_to_lds_b128(
      (gv4i*)(unsigned long long)g,
      (lv4i*)(unsigned int)(unsigned long long)l,
      0, 0);
}
#else
#define USE_ASYNC 0
#endif

__device__ __forceinline__ float fast_rcp(float a) {
#if __has_builtin(__builtin_amdgcn_rcpf)
  return __builtin_amdgcn_rcpf(a);   // single v_rcp_f32
#else
  return 1.0f / a;
#endif
}

__global__ void zero_counts_kernel(float* __restrict__ cnt) {
  if (threadIdx.x < NE) cnt[threadIdx.x] = 0.0f;
}

__global__ __launch_bounds__(128) void router_kernel(
    const float* __restrict__ x,        // (T, DIM)
    const float* __restrict__ gw,       // (NE, DIM)
    const float* __restrict__ bias,     // (NE)
    float* __restrict__ out_scores,     // (T, TOPK)
    float* __restrict__ out_idx,        // (T, TOPK) expert ids as float
    float* __restrict__ out_cnt) {      // (NE)
  // Double-buffered GEMM staging; score tables overlaid after the last barrier.
  union SMem {
    struct { float xs[2][BT][XROW]; float gs[2][NE][XROW]; } a;
    struct { float sc[BT][SROW];    float bs[BT][SROW];    } b;
  };
  __shared__ SMem sm;
  __shared__ float sbias[NE];
  __shared__ int   hist[NE];

  const int tid  = threadIdx.x;
  const int lane = tid & 31;
  const int wave = tid >> 5;            // 4 waves; wave w owns tokens [16w, 16w+16)
  const int t0   = blockIdx.x * BT;

  if (tid < NE) { hist[tid] = 0; sbias[tid] = bias[tid]; }

  v8f acc[4] = {};                      // 16x16 f32 C tiles for expert groups 0..3

  const int m  = lane & 15;             // fragment row (M for A, N for B)
  const int kh = (lane >> 4) << 1;      // lanes 0-15 -> K{0,1}; lanes 16-31 -> K{2,3}

  // Stage one 128-wide K chunk of x (64 rows) and gate_w (64 rows) into buffer `buf`.
  auto stage_chunk = [&](int buf, int kb) {
#pragma unroll
    for (int i = 0; i < 16; i++) {
      int idx = tid + (i << 7);         // 0..2047 float4 slots
      int r = idx >> 5, c = (idx & 31) << 2;
#if USE_ASYNC
      async_copy_b128(x + (size_t)(t0 + r) * DIM + kb + c, &sm.a.xs[buf][r][c]);
#else
      *(float4*)&sm.a.xs[buf][r][c] =
          *(const float4*)(x + (size_t)(t0 + r) * DIM + kb + c);
#endif
    }
#pragma unroll
    for (int i = 0; i < 16; i++) {
      int idx = tid + (i << 7);
      int r = idx >> 5, c = (idx & 31) << 2;
#if USE_ASYNC
      async_copy_b128(gw + (size_t)r * DIM + kb + c, &sm.a.gs[buf][r][c]);
#else
      *(float4*)&sm.a.gs[buf][r][c] =
          *(const float4*)(gw + (size_t)r * DIM + kb + c);
#endif
    }
  };

  stage_chunk(0, 0);                    // prologue: fill buffer 0
#if USE_ASYNC
  asm volatile("s_wait_asynccnt 0" ::: "memory");
#endif
  __syncthreads();

  int cur = 0;
  for (int it = 0; it < DIM / KC; ++it) {
    const int kb_next = (it + 1) * KC;
    if (kb_next < DIM) stage_chunk(cur ^ 1, kb_next);  // overlap with WMMA below

    // fp32 WMMA over the resident chunk: 1 A load reused by 4 expert tiles
#pragma unroll 4
    for (int kk = 0; kk < KC; kk += 4) {
      v2f a = *(const v2f*)&sm.a.xs[cur][(wave << 4) + m][kk + kh];
#pragma unroll
      for (int j = 0; j < 4; j++) {
        v2f b = *(const v2f*)&sm.a.gs[cur][(j << 4) + m][kk + kh];
        acc[j] = __builtin_amdgcn_wmma_f32_16x16x4_f32(
            false, a, false, b, (short)0, acc[j], false, false);
      }
    }
#if USE_ASYNC
    asm volatile("s_wait_asynccnt 0" ::: "memory");     // this wave's copies done
#endif
    __syncthreads();                                    // all waves' copies visible
    cur ^= 1;
  }

  // ---- sigmoid (fp32, single v_rcp) + scatter into overlaid LDS score tables ----
#pragma unroll
  for (int j = 0; j < 4; j++) {
#pragma unroll
    for (int i = 0; i < 8; i++) {
      float z = acc[j][i];
      float s = fast_rcp(1.0f + __expf(-z));
      int tok = (wave << 4) + i + ((lane >> 4) << 3);   // C layout: VGPR i -> M=i / i+8
      int e   = (j << 4) + m;
      sm.b.sc[tok][e] = s;
      sm.b.bs[tok][e] = s + sbias[e];
    }
  }
  __syncthreads();

  // ---- top-8 with knockout; '>' keeps lowest index on ties (matches lax.top_k) ----
  if (tid < BT) {
    const int tok = tid;
#pragma unroll 1
    for (int k = 0; k < TOPK; k++) {
      float best = -INFINITY; int be = 0;
      for (int e = 0; e < NE; e++) {
        float v = sm.b.bs[tok][e];
        if (v > best) { best = v; be = e; }
      }
      sm.b.bs[tok][be] = -INFINITY;
      size_t o = (size_t)(t0 + tok) * TOPK + k;
      out_scores[o] = sm.b.sc[tok][be];
      out_idx[o]    = (float)be;
      atomicAdd(&hist[be], 1);
    }
  }
  __syncthreads();

  // ---- one global float atomic per expert per block (exact integer sums) ----
  if (tid < NE) {
    int c = hist[tid];
    if (c) atomicAdd(&out_cnt[tid], (float)c);
  }
}

extern "C" void kernel_launch(void* const* d_in, const int* in_sizes, int n_in,
                              void* d_out, int out_size, void* d_ws, size_t ws_size,
                              hipStream_t stream) {
  (void)in_sizes; (void)n_in; (void)out_size; (void)d_ws; (void)ws_size;
  const float* x    = (const float*)d_in[0];
  const float* gw   = (const float*)d_in[1];
  const float* bias = (const float*)d_in[2];
  float* out        = (float*)d_out;
  float* out_scores = out;
  float* out_idx    = out + (size_t)T_TOKENS * TOPK;
  float* out_cnt    = out + (size_t)2 * T_TOKENS * TOPK;

  zero_counts_kernel<<<1, 64, 0, stream>>>(out_cnt);
  router_kernel<<<T_TOKENS / BT, 128, 0, stream>>>(x, gw, bias, out_scores, out_idx, out_cnt);
}